// Equivariance_84430467104992
// MI455X (gfx1250) — compile-verified
//
#include <hip/hip_runtime.h>
#include <hip/hip_bf16.h>

#define NCONF 8192
#define NT 64
#define NX 64
#define LDP 65   // padded LDS row stride (dwords) to avoid 64-bank conflicts

typedef __attribute__((ext_vector_type(2))) float v2f;
typedef __attribute__((ext_vector_type(4))) float v4f;
typedef __attribute__((ext_vector_type(8))) float v8f;

// ---------------------------------------------------------------------------
// Kernel 1: log|det(W)| via Gaussian elimination in LDS (W = I + 0.1*N is
// well-conditioned; no pivoting required for the value). Writes 64*log|det|.
// ---------------------------------------------------------------------------
__global__ __launch_bounds__(64) void logdet_kernel(const float* __restrict__ W,
                                                    float* __restrict__ scratch) {
    __shared__ float A[NX * NX];
    __shared__ float logsum;
    const int tid = threadIdx.x;  // 64 threads = 2 waves

    for (int e = tid; e < NX * NX; e += 64) A[e] = W[e];
    if (tid == 0) logsum = 0.0f;
    __syncthreads();

    for (int k = 0; k < NX; ++k) {
        const float piv = A[k * NX + k];
        if (tid == 0) logsum += __logf(fabsf(piv));
        if (tid > k) {
            const float f = A[tid * NX + k] / piv;
            for (int j = k + 1; j < NX; ++j)
                A[tid * NX + j] -= f * A[k * NX + j];
        }
        __syncthreads();
    }
    if (tid == 0) scratch[0] = (float)NT * logsum;
}

// ---------------------------------------------------------------------------
// Kernel 2: broadcast logDet scalar to the 8192-element tail of d_out.
// ---------------------------------------------------------------------------
__global__ void fill_logdet_kernel(const float* __restrict__ scratch,
                                   float* __restrict__ outLog, int n) {
    const int i = blockIdx.x * blockDim.x + threadIdx.x;
    if (i < n) outLog[i] = scratch[0];
}

// ---------------------------------------------------------------------------
// Kernel 3: per-config shift detection + 64x64x64 f32 GEMM via WMMA.
//
// Reference folds to:  out[i,t,j] = sum_u phi[t,u] * W[(u+T0)&63,(j+T0)&63]
//                                   + b[(j+T0)&63]
// so we stage phi STRAIGHT (coalesced, no T0 dependency) and roll W once
// during its LDS staging. Inner loop then has only static-immediate LDS
// reads (ds_load_2addr for both A and B fragments) feeding
// v_wmma_f32_16x16x4_f32, and the output stores are fully coalesced.
// ---------------------------------------------------------------------------
__global__ __launch_bounds__(256) void equiv_gemm_kernel(
    const float* __restrict__ phi, const float* __restrict__ W,
    const float* __restrict__ bvec, float* __restrict__ out) {
    __shared__ float sPhi[NT * LDP];   // 64 x 65 (padded), straight copy
    __shared__ float sW[NX * LDP];     // 64 x 65 (padded), rolled by T0
    __shared__ unsigned long long sKey[2];
    __shared__ int sT0;

    const int conf = blockIdx.x;
    const int tid  = threadIdx.x;
    const float* __restrict__ phiC = phi + (size_t)conf * (NT * NX);

    if (tid == 0) { sKey[0] = ~0ull; sKey[1] = ~0ull; }

    // argmin keys straight from global (independent of phi staging):
    // key = bits(|v|) << 6 | t  -> first-index tie-break like jnp.argmin.
    if (tid < NT) {
        const unsigned a0 = __float_as_uint(fabsf(phiC[tid * NX + 0]));
        const unsigned a1 = __float_as_uint(fabsf(phiC[tid * NX + 1]));
        atomicMin(&sKey[0], (((unsigned long long)a0) << 6) | (unsigned)tid);
        atomicMin(&sKey[1], (((unsigned long long)a1) << 6) | (unsigned)tid);
    }

    // Stage phi tile straight into LDS: coalesced non-temporal b128 loads
    // (phi is streamed exactly once; don't pollute L2 needed for W/out).
    const v4f* __restrict__ phiC4 = (const v4f*)phiC;
    #pragma unroll
    for (int k = 0; k < (NT * NX) / (4 * 256); ++k) {
        const int i4 = tid + k * 256;
        const v4f v = __builtin_nontemporal_load(&phiC4[i4]);
        const int e = i4 * 4;
        const int r = e >> 6, c = e & 63;   // c is a multiple of 4
        float* dst = &sPhi[r * LDP + c];
        dst[0] = v.x; dst[1] = v.y; dst[2] = v.z; dst[3] = v.w;
    }
    __syncthreads();

    if (tid == 0) {
        const int t0 = (int)(sKey[0] & 63u);
        const int t1 = (int)(sKey[1] & 63u);
        sT0 = (t0 > t1) ? t1 : t0;
    }
    __syncthreads();
    const int T0 = sT0;

    // Stage W rolled by T0 in both dims (one-time; W is 16 KB and L2-hot).
    for (int e = tid; e < NX * NX; e += 256) {
        const int u = e >> 6, y = e & 63;
        sW[u * LDP + y] = W[((u + T0) & 63) * NX + ((y + T0) & 63)];
    }
    __syncthreads();

    // Tile assignment: 4x4 grid of 16x16 tiles; wave w -> row tm = w&3,
    // column pair tn = {2*(w>>2), 2*(w>>2)+1} (A fragment reused for both).
    const int wave   = tid >> 5;
    const int lane   = tid & 31;
    const int lh     = lane & 15;
    const int hi     = lane >> 4;          // lane half: K offset 0 or 2
    const int tm     = wave & 3;
    const int tnBase = (wave >> 2) * 2;

    const int rowA = tm * 16 + lh;         // A row (t); same for both halves
    const int n0   = tnBase * 16 + lh;     // B/D column for tile 0
    const int n1   = n0 + 16;              // B/D column for tile 1

    v8f c0 = {};
    v8f c1 = {};
    #pragma unroll
    for (int k0 = 0; k0 < NX; k0 += 4) {
        const int ka = k0 + hi * 2;        // .x holds K=ka, .y holds K=ka+1
        v2f a, b0, b1;
        a.x  = sPhi[rowA * LDP + ka];      // static offsets -> ds_load_2addr
        a.y  = sPhi[rowA * LDP + ka + 1];
        b0.x = sW[ka * LDP + n0];
        b0.y = sW[(ka + 1) * LDP + n0];
        b1.x = sW[ka * LDP + n1];
        b1.y = sW[(ka + 1) * LDP + n1];
        c0 = __builtin_amdgcn_wmma_f32_16x16x4_f32(false, a, false, b0,
                                                   (short)0, c0, false, false);
        c1 = __builtin_amdgcn_wmma_f32_16x16x4_f32(false, a, false, b1,
                                                   (short)0, c1, false, false);
    }

    // Epilogue: rolled bias, fully coalesced non-temporal stores.
    float* __restrict__ outC = out + (size_t)conf * (NT * NX);
    const float bv0 = bvec[(n0 + T0) & 63];
    const float bv1 = bvec[(n1 + T0) & 63];
    #pragma unroll
    for (int v = 0; v < 8; ++v) {
        const int row = tm * 16 + v + hi * 8;   // C/D: VGPR v -> M = v (+8 hi)
        __builtin_nontemporal_store(c0[v] + bv0, &outC[row * NX + n0]);
        __builtin_nontemporal_store(c1[v] + bv1, &outC[row * NX + n1]);
    }
}

// ---------------------------------------------------------------------------
extern "C" void kernel_launch(void* const* d_in, const int* in_sizes, int n_in,
                              void* d_out, int out_size, void* d_ws, size_t ws_size,
                              hipStream_t stream) {
    const float* phi  = (const float*)d_in[0];
    const float* W    = (const float*)d_in[1];
    const float* bvec = (const float*)d_in[2];
    float* out        = (float*)d_out;
    float* scratch    = (float*)d_ws;

    logdet_kernel<<<1, 64, 0, stream>>>(W, scratch);
    fill_logdet_kernel<<<(NCONF + 255) / 256, 256, 0, stream>>>(
        scratch, out + (size_t)NCONF * NT * NX, NCONF);
    equiv_gemm_kernel<<<NCONF, 256, 0, stream>>>(phi, W, bvec, out);
}